// SGPNOutput_27084063768631
// MI455X (gfx1250) — compile-verified
//
#include <hip/hip_runtime.h>
#include <cstddef>

#define BB 4
#define NN 4096
#define CC 256
#define DD 128
#define GG 50

typedef __attribute__((ext_vector_type(2))) float v2f;
typedef __attribute__((ext_vector_type(8))) float v8f;

// ---------------------------------------------------------------------------
// Phase 1: per-row feature heads. One 128-thread block per (b,n) row.
//  - Fsem -> seg logits -> softmax   (outputs 0,1)
//  - Fsim -> workspace, r = sum(Fsim^2) -> workspace
//  - Fconf -> conf logit -> sigmoid  (outputs 3,4)
// ---------------------------------------------------------------------------
__global__ __launch_bounds__(128) void features_kernel(
    const float* __restrict__ x,
    const float* __restrict__ W_sem, const float* __restrict__ b_sem,
    const float* __restrict__ W_seg, const float* __restrict__ b_seg,
    const float* __restrict__ W_sim, const float* __restrict__ b_sim,
    const float* __restrict__ W_conf, const float* __restrict__ b_conf,
    const float* __restrict__ W_cl,  const float* __restrict__ b_cl,
    float* __restrict__ o_soft, float* __restrict__ o_slog,
    float* __restrict__ o_conf, float* __restrict__ o_clog,
    float* __restrict__ wsF, float* __restrict__ wsR)
{
    __shared__ float xs[CC];
    __shared__ float fs[DD];
    __shared__ float sg[64];
    __shared__ float es[64];

    const int t  = threadIdx.x;
    const int bn = blockIdx.x;
    const float* xr = x + (size_t)bn * CC;
    xs[t]      = xr[t];
    xs[t + DD] = xr[t + DD];
    __syncthreads();

    // ---- semantic head: Fsem[d] = x . W_sem[:,d] + b  (coalesced W columns)
    float a = b_sem[t];
    #pragma unroll 4
    for (int c = 0; c < CC; ++c) a = fmaf(xs[c], W_sem[c * DD + t], a);
    fs[t] = a;
    __syncthreads();

    if (t < GG) {
        float lg = b_seg[t];
        #pragma unroll 4
        for (int d = 0; d < DD; ++d) lg = fmaf(fs[d], W_seg[d * GG + t], lg);
        sg[t] = lg;
        o_slog[(size_t)bn * GG + t] = lg;
    }
    __syncthreads();
    if (t < GG) {
        float mx = -3.402823466e38f;
        for (int j = 0; j < GG; ++j) mx = fmaxf(mx, sg[j]);
        es[t] = __expf(sg[t] - mx);
    }
    __syncthreads();
    if (t < GG) {
        float s = 0.f;
        for (int j = 0; j < GG; ++j) s += es[j];
        o_soft[(size_t)bn * GG + t] = es[t] / s;
    }
    __syncthreads();

    // ---- similarity head: Fsim -> workspace (L2-resident for phase 2)
    float v = b_sim[t];
    #pragma unroll 4
    for (int c = 0; c < CC; ++c) v = fmaf(xs[c], W_sim[c * DD + t], v);
    wsF[(size_t)bn * DD + t] = v;
    fs[t] = v * v;
    __syncthreads();
    for (int s = 64; s > 0; s >>= 1) {
        if (t < s) fs[t] += fs[t + s];
        __syncthreads();
    }
    if (t == 0) wsR[bn] = fs[0];
    __syncthreads();

    // ---- confidence head: Fconf . W_cl + b_cl -> sigmoid
    float q = b_conf[t];
    #pragma unroll 4
    for (int c = 0; c < CC; ++c) q = fmaf(xs[c], W_conf[c * DD + t], q);
    fs[t] = q * W_cl[t];
    __syncthreads();
    for (int s = 64; s > 0; s >>= 1) {
        if (t < s) fs[t] += fs[t + s];
        __syncthreads();
    }
    if (t == 0) {
        float cl = fs[0] + b_cl[0];
        o_clog[bn] = cl;
        o_conf[bn] = 1.0f / (1.0f + __expf(-cl));
    }
}

// ---------------------------------------------------------------------------
// Phase 2: gram = Fsim @ Fsim^T per batch via V_WMMA_F32_16X16X4_F32, fused
// with simmat = relu(10*(r_n - 2*gram + r_m)).
// simmat is SYMMETRIC: only upper-triangular 64x64 tiles (tm >= tn) are
// computed (halves the fp32 WMMA work, which dominates over the 11.5us HBM
// store floor). Off-diagonal tiles stage the result transposed in LDS and
// also emit the coalesced mirror tile, so total stores are unchanged.
// Block = 128 threads (4 wave32); each wave owns a 16x64 strip of the tile
// (4 x 16x16 f32 accumulators -> no WMMA RAW hazards). K=128 via 2 LDS chunks.
// ---------------------------------------------------------------------------
__global__ __launch_bounds__(128) void gram_kernel(
    const float* __restrict__ F, const float* __restrict__ R,
    float* __restrict__ S)
{
    const int tn = blockIdx.x;          // n-tile (rows)
    const int tm = blockIdx.y;          // m-tile (cols)
    if (tm < tn) return;                // uniform exit: symmetry

    __shared__ __align__(16) float As[64][68];   // stride 68: bank stride 4, rows 16B-aligned
    __shared__ __align__(16) float Bs[64][68];
    __shared__ float rA[64];
    __shared__ float rB[64];

    const int t  = threadIdx.x;
    const int n0 = tn * 64;
    const int m0 = tm * 64;
    const int b  = blockIdx.z;
    const float* Fb = F + (size_t)b * NN * DD;

    if (t < 64) rA[t]      = R[(size_t)b * NN + n0 + t];
    else        rB[t - 64] = R[(size_t)b * NN + m0 + (t - 64)];

    const int l     = t & 31;   // lane in wave32
    const int w     = t >> 5;   // wave id 0..3
    const int lrow  = l & 15;
    const int lhalf = l >> 4;

    v8f acc[4] = {};

    for (int kc = 0; kc < DD; kc += 64) {
        __syncthreads();
        // fill LDS: 64 rows x 64 K-floats per tile, float4 per thread x 8
        #pragma unroll
        for (int it = 0; it < 8; ++it) {
            int idx = it * 128 + t;
            int row = idx >> 4;
            int cv  = (idx & 15) << 2;
            *(float4*)&As[row][cv] =
                *(const float4*)&Fb[(size_t)(n0 + row) * DD + kc + cv];
            *(float4*)&Bs[row][cv] =
                *(const float4*)&Fb[(size_t)(m0 + row) * DD + kc + cv];
        }
        __syncthreads();

        for (int kk = 0; kk < 64; kk += 4) {
            const int ka = kk + 2 * lhalf;   // lanes 0-15: K=kk,kk+1 ; lanes 16-31: K=kk+2,kk+3
            v2f av;
            av.x = As[w * 16 + lrow][ka];
            av.y = As[w * 16 + lrow][ka + 1];
            #pragma unroll
            for (int j = 0; j < 4; ++j) {
                v2f bv;
                bv.x = Bs[j * 16 + lrow][ka];
                bv.y = Bs[j * 16 + lrow][ka + 1];
                // D(16x16,f32) += A(16x4,f32) * B(4x16,f32)
                acc[j] = __builtin_amdgcn_wmma_f32_16x16x4_f32(
                    false, av, false, bv, (short)0, acc[j], false, false);
            }
        }
    }
    __syncthreads();

    // epilogue: val = relu(10*(r_n - 2*gram + r_m)); direct store + stage
    // the transpose in LDS (reuse As) for the mirror tile.
    #pragma unroll
    for (int j = 0; j < 4; ++j) {
        #pragma unroll
        for (int v = 0; v < 8; ++v) {
            int nn = w * 16 + v + 8 * lhalf;   // C-layout: VGPR v -> M = v + 8*half
            int mm = j * 16 + lrow;            //            lane%16 -> N
            float Dv  = rA[nn] - 2.0f * acc[j][v] + rB[mm];
            float val = fmaxf(10.0f * Dv, 0.0f);
            S[((size_t)b * NN + (n0 + nn)) * NN + (m0 + mm)] = val;
            As[mm][nn] = val;                  // transposed stage
        }
    }
    if (tn != tm) {
        __syncthreads();
        // mirror tile: rows m0..m0+63, cols n0..n0+63, coalesced float4 rows
        #pragma unroll
        for (int it = 0; it < 8; ++it) {
            int idx = it * 128 + t;
            int row = idx >> 4;
            int cv  = (idx & 15) << 2;
            *(float4*)&S[((size_t)b * NN + (m0 + row)) * NN + n0 + cv] =
                *(const float4*)&As[row][cv];
        }
    }
}

// ---------------------------------------------------------------------------
extern "C" void kernel_launch(void* const* d_in, const int* in_sizes, int n_in,
                              void* d_out, int out_size, void* d_ws, size_t ws_size,
                              hipStream_t stream) {
    const float* x      = (const float*)d_in[0];
    const float* W_sem  = (const float*)d_in[1];
    const float* b_sem  = (const float*)d_in[2];
    const float* W_seg  = (const float*)d_in[3];
    const float* b_seg  = (const float*)d_in[4];
    const float* W_sim  = (const float*)d_in[5];
    const float* b_sim  = (const float*)d_in[6];
    const float* W_conf = (const float*)d_in[7];
    const float* b_conf = (const float*)d_in[8];
    const float* W_cl   = (const float*)d_in[9];
    const float* b_cl   = (const float*)d_in[10];

    float* O      = (float*)d_out;
    float* o_soft = O;
    float* o_slog = O + (size_t)BB * NN * GG;
    float* o_sim  = o_slog + (size_t)BB * NN * GG;
    float* o_conf = o_sim + (size_t)BB * NN * NN;
    float* o_clog = o_conf + (size_t)BB * NN;

    float* wsF = (float*)d_ws;                       // 4*4096*128 f32 = 8 MB
    float* wsR = wsF + (size_t)BB * NN * DD;         // + 16384 f32

    features_kernel<<<BB * NN, 128, 0, stream>>>(
        x, W_sem, b_sem, W_seg, b_seg, W_sim, b_sim, W_conf, b_conf, W_cl, b_cl,
        o_soft, o_slog, o_conf, o_clog, wsF, wsR);

    dim3 grid(NN / 64, NN / 64, BB);
    gram_kernel<<<grid, 128, 0, stream>>>(wsF, wsR, o_sim);
}